// CCQAAttention_11192684773439
// MI455X (gfx1250) — compile-verified
//
#include <hip/hip_runtime.h>

#define BDIM 4
#define SEQ 2048
#define DIMX 2048
#define LATD 512
#define NHQ 16
#define NKVH 4
#define HDIM 32
#define KVDIM 128
#define GAINF 0.25f
#define EPSF 1e-6f
#define ROWS (BDIM * SEQ)

typedef float v2f __attribute__((ext_vector_type(2)));
typedef float v8f __attribute__((ext_vector_type(8)));

__device__ __forceinline__ v8f wmma_f32(v2f a, v2f b, v8f c) {
  // D = A(16x4) * B(4x16) + C, fp32 exact
  return __builtin_amdgcn_wmma_f32_16x16x4_f32(false, a, false, b, (short)0, c,
                                               false, false);
}

// ---------------------------------------------------------------------------
// C[M x N] = A[M x K] * B[N x K]^T  (row-major, B is a weight (N, K))
// One wave per 32(M) x 64(N) tile. grid = (N/64, M/32), block = 32.
// K processed in chunks of 16: 24 batched b64 loads, then 32 WMMAs.
// ---------------------------------------------------------------------------
__global__ __launch_bounds__(32) void gemm_nt(const float* __restrict__ A,
                                              const float* __restrict__ B,
                                              float* __restrict__ C,
                                              int N, int Kd) {
  const int lane = threadIdx.x;
  const int half = lane >> 4;
  const int lm = lane & 15;
  const int n0 = blockIdx.x * 64;
  const int m0 = blockIdx.y * 32;

  const float* a0r = A + (size_t)(m0 + lm) * Kd;
  const float* a1r = a0r + (size_t)16 * Kd;
  const float* b0r = B + (size_t)(n0 + lm) * Kd;
  const float* b1r = b0r + (size_t)16 * Kd;
  const float* b2r = b0r + (size_t)32 * Kd;
  const float* b3r = b0r + (size_t)48 * Kd;

  v8f c00 = {}, c01 = {}, c02 = {}, c03 = {};
  v8f c10 = {}, c11 = {}, c12 = {}, c13 = {};

  for (int k0 = 0; k0 < Kd; k0 += 16) {
    v2f a0[4], a1[4], bb0[4], bb1[4], bb2[4], bb3[4];
#pragma unroll
    for (int j = 0; j < 4; ++j) {
      const int kk = k0 + 4 * j + 2 * half;
      a0[j] = *(const v2f*)(a0r + kk);
      a1[j] = *(const v2f*)(a1r + kk);
      bb0[j] = *(const v2f*)(b0r + kk);
      bb1[j] = *(const v2f*)(b1r + kk);
      bb2[j] = *(const v2f*)(b2r + kk);
      bb3[j] = *(const v2f*)(b3r + kk);
    }
#pragma unroll
    for (int j = 0; j < 4; ++j) {
      c00 = wmma_f32(a0[j], bb0[j], c00);
      c01 = wmma_f32(a0[j], bb1[j], c01);
      c02 = wmma_f32(a0[j], bb2[j], c02);
      c03 = wmma_f32(a0[j], bb3[j], c03);
      c10 = wmma_f32(a1[j], bb0[j], c10);
      c11 = wmma_f32(a1[j], bb1[j], c11);
      c12 = wmma_f32(a1[j], bb2[j], c12);
      c13 = wmma_f32(a1[j], bb3[j], c13);
    }
  }

  float* crow0 = C + (size_t)(m0 + 8 * half) * N + n0 + lm;
  float* crow1 = crow0 + (size_t)16 * N;
#pragma unroll
  for (int r = 0; r < 8; ++r) {
    float* p0 = crow0 + (size_t)r * N;
    float* p1 = crow1 + (size_t)r * N;
    p0[0] = c00[r];
    p0[16] = c01[r];
    p0[32] = c02[r];
    p0[48] = c03[r];
    p1[0] = c10[r];
    p1[16] = c11[r];
    p1[32] = c12[r];
    p1[48] = c13[r];
  }
}

// ---------------------------------------------------------------------------
// RMS norm over rows: one wave per row, in-place safe (values register-cached)
// ---------------------------------------------------------------------------
__global__ __launch_bounds__(32) void rms_rows(const float* __restrict__ in,
                                               int stride,
                                               const float* __restrict__ g,
                                               float* __restrict__ out,
                                               int len) {
  const int row = blockIdx.x;
  const int lane = threadIdx.x;
  const float* src = in + (size_t)row * stride;
  float xv[16];
  const int nv = len >> 5;
  float ss = 0.f;
  for (int i = 0; i < nv; ++i) {
    float x = src[lane + (i << 5)];
    xv[i] = x;
    ss += x * x;
  }
#pragma unroll
  for (int m = 16; m >= 1; m >>= 1) ss += __shfl_xor(ss, m);
  const float rinv = rsqrtf(ss / (float)len + EPSF);
  float* dst = out + (size_t)row * len;
  for (int i = 0; i < nv; ++i)
    dst[lane + (i << 5)] = xv[i] * rinv * g[lane + (i << 5)];
}

// ---------------------------------------------------------------------------
// Grouped causal conv, K=3, 32 channels per group (both q and k convs).
// in: (ROWS, C) compact; w: (C, 32, 3); out: (ROWS, C)
// ---------------------------------------------------------------------------
__global__ __launch_bounds__(256) void conv_causal(const float* __restrict__ in,
                                                   const float* __restrict__ w,
                                                   float* __restrict__ out,
                                                   int C, int total) {
  const int idx = blockIdx.x * 256 + threadIdx.x;
  if (idx >= total) return;
  const int o = idx % C;
  const int row = idx / C;
  const int s = row & (SEQ - 1);
  const int gb = o & ~31;
  const float* wr = w + (size_t)o * 96;
  const float* base = in + (size_t)row * C + gb;
  float acc = 0.f;
#pragma unroll 4
  for (int c = 0; c < 32; ++c) {
    acc = fmaf(wr[c * 3 + 2], base[c], acc);
    if (s >= 1) acc = fmaf(wr[c * 3 + 1], base[c - C], acc);
    if (s >= 2) acc = fmaf(wr[c * 3 + 0], base[c - 2 * C], acc);
  }
  out[(size_t)row * C + o] = acc;
}

// ---------------------------------------------------------------------------
// q finalize: += GAIN*mean_kv(k_pre), RMS(g_postq), per-head L2*sqrt(HD), RoPE
// block = 512 (one wave == one head), grid = ROWS. In-place safe.
// ---------------------------------------------------------------------------
__global__ __launch_bounds__(512) void finalize_q(const float* __restrict__ qc,
                                                  const float* __restrict__ kpre,
                                                  const float* __restrict__ gpost,
                                                  float* __restrict__ qout) {
  const int row = blockIdx.x;
  const int t = threadIdx.x;
  const int s = row & (SEQ - 1);
  const int hd = t & 31;

  const float* kr = kpre + (size_t)row * KVDIM;
  float km = 0.f;
#pragma unroll
  for (int j = 0; j < NKVH; ++j) km += kr[j * HDIM + hd];
  float val = qc[(size_t)row * LATD + t] + GAINF * (km * (1.f / NKVH));

  __shared__ float red[16];
  float ss = val * val;
#pragma unroll
  for (int m = 16; m >= 1; m >>= 1) ss += __shfl_xor(ss, m);
  if ((t & 31) == 0) red[t >> 5] = ss;
  __syncthreads();
  float tot = 0.f;
#pragma unroll
  for (int i = 0; i < 16; ++i) tot += red[i];
  val = val * rsqrtf(tot / (float)LATD + EPSF) * gpost[t];

  float nn = val * val;
#pragma unroll
  for (int m = 16; m >= 1; m >>= 1) nn += __shfl_xor(nn, m);
  val = val / fmaxf(sqrtf(nn), 1e-12f) * 5.65685424949238f;  // sqrt(32)

  const int i2 = hd >> 1;
  const float invf = expf(-(float)(2 * i2) * (9.210340371976184f / 32.f));
  const float f = (float)s * invf;
  const float cs = cosf(f), sn = sinf(f);
  const float other = __shfl_xor(val, 1);
  const float r = (hd & 1) ? (other * sn + val * cs) : (val * cs - other * sn);
  qout[(size_t)row * LATD + t] = r;
}

// ---------------------------------------------------------------------------
// k finalize: += GAIN*mean_h(q_pre), RMS(g_postk), L2*sqrt(HD)*key_temp, RoPE
// block = 128 (one wave == one kv head), grid = ROWS. In-place safe.
// ---------------------------------------------------------------------------
__global__ __launch_bounds__(128) void finalize_k(const float* __restrict__ kc,
                                                  const float* __restrict__ qpre,
                                                  const float* __restrict__ gpost,
                                                  const float* __restrict__ ktmp,
                                                  float* __restrict__ kout) {
  const int row = blockIdx.x;
  const int t = threadIdx.x;
  const int s = row & (SEQ - 1);
  const int hd = t & 31;

  const float* qr = qpre + (size_t)row * LATD;
  float qm = 0.f;
#pragma unroll
  for (int j = 0; j < NHQ; ++j) qm += qr[j * HDIM + hd];
  float val = kc[(size_t)row * KVDIM + t] + GAINF * (qm * (1.f / NHQ));

  __shared__ float red[4];
  float ss = val * val;
#pragma unroll
  for (int m = 16; m >= 1; m >>= 1) ss += __shfl_xor(ss, m);
  if ((t & 31) == 0) red[t >> 5] = ss;
  __syncthreads();
  float tot = red[0] + red[1] + red[2] + red[3];
  val = val * rsqrtf(tot / (float)KVDIM + EPSF) * gpost[t];

  float nn = val * val;
#pragma unroll
  for (int m = 16; m >= 1; m >>= 1) nn += __shfl_xor(nn, m);
  val = val / fmaxf(sqrtf(nn), 1e-12f) * 5.65685424949238f * ktmp[0];

  const int i2 = hd >> 1;
  const float invf = expf(-(float)(2 * i2) * (9.210340371976184f / 32.f));
  const float f = (float)s * invf;
  const float cs = cosf(f), sn = sinf(f);
  const float other = __shfl_xor(val, 1);
  const float r = (hd & 1) ? (other * sn + val * cs) : (val * cs - other * sn);
  kout[(size_t)row * KVDIM + t] = r;
}

// ---------------------------------------------------------------------------
// Flash attention: one wave per (b, h, 16-query tile). Causal, GQA (kh = h/4).
// Scores and PV both via fp32 WMMA. V operands prefetched before the score
// WMMA burst so their latency hides under QK^T. grid = (SEQ/16, NH, B).
// ---------------------------------------------------------------------------
__global__ __launch_bounds__(32) void attn_kernel(const float* __restrict__ q,
                                                  const float* __restrict__ k,
                                                  const float* __restrict__ v,
                                                  float* __restrict__ o) {
  const int lane = threadIdx.x;
  const int half = lane >> 4;
  const int lm = lane & 15;
  const int qt = blockIdx.x;
  const int h = blockIdx.y;
  const int b = blockIdx.z;
  const int kh = h >> 2;
  const int qs0 = qt * 16;

  __shared__ float pb[16 * 17];

  // Preload Q tile (16 x 32): A-operand layout, 8 chunks of K=4
  const float* qrow = q + (size_t)(b * SEQ + qs0 + lm) * LATD + h * HDIM;
  v2f aq[8];
#pragma unroll
  for (int j = 0; j < 8; ++j)
    aq[j] = *(const v2f*)(qrow + 4 * j + 2 * half);

  v8f acc0 = {}, acc1 = {};
  float mrow[8], lrow[8];
#pragma unroll
  for (int r = 0; r < 8; ++r) {
    mrow[r] = -1e30f;
    lrow[r] = 0.f;
  }

  const float* kb = k + (size_t)(b * SEQ) * KVDIM + kh * HDIM;
  const float* vb = v + (size_t)(b * SEQ) * KVDIM + kh * HDIM;
  const float scale = 0.17677669529663687f;  // 1/sqrt(32)

  for (int kt = 0; kt <= qt; ++kt) {
    const int ks0 = kt * 16;
    const float* krow = kb + (size_t)(ks0 + lm) * KVDIM;

    // Batch K-tile loads (8 x b64)
    v2f bk[8];
#pragma unroll
    for (int j = 0; j < 8; ++j)
      bk[j] = *(const v2f*)(krow + 4 * j + 2 * half);

    // Prefetch V-tile B-operands (independent of scores)
    v2f vb0[4], vb1[4];
#pragma unroll
    for (int s4 = 0; s4 < 4; ++s4) {
      const int key0 = 4 * s4 + 2 * half;
      const float* vr0 = vb + (size_t)(ks0 + key0) * KVDIM;
      const float* vr1 = vr0 + KVDIM;
      vb0[s4].x = vr0[lm];
      vb0[s4].y = vr1[lm];
      vb1[s4].x = vr0[lm + 16];
      vb1[s4].y = vr1[lm + 16];
    }

    // scores 16x16 = Q(16x32) . K^T(32x16)
    v8f c = {};
#pragma unroll
    for (int j = 0; j < 8; ++j) c = wmma_f32(aq[j], bk[j], c);

    // mask + online softmax (row r+8*half lives across 16 lanes of this half)
#pragma unroll
    for (int r = 0; r < 8; ++r) {
      float sc = c[r] * scale;
      const int qi = qs0 + r + 8 * half;
      const int ki = ks0 + lm;
      if (ki > qi) sc = -1e30f;
      float mx = sc;
#pragma unroll
      for (int mm = 1; mm < 16; mm <<= 1) mx = fmaxf(mx, __shfl_xor(mx, mm));
      const float mnew = fmaxf(mrow[r], mx);
      const float alpha = __expf(mrow[r] - mnew);
      const float p = __expf(sc - mnew);
      float rs = p;
#pragma unroll
      for (int mm = 1; mm < 16; mm <<= 1) rs += __shfl_xor(rs, mm);
      lrow[r] = lrow[r] * alpha + rs;
      mrow[r] = mnew;
      acc0[r] *= alpha;
      acc1[r] *= alpha;
      pb[(r + 8 * half) * 17 + lm] = p;
    }
    __syncthreads();

    // Batch LDS reads of transposed P (A-operand layout)
    v2f ap[4];
#pragma unroll
    for (int s4 = 0; s4 < 4; ++s4) {
      const int key0 = 4 * s4 + 2 * half;
      ap[s4].x = pb[lm * 17 + key0];
      ap[s4].y = pb[lm * 17 + key0 + 1];
    }

    // PV: out(16x32) += P(16x16) . V(16x32), K in chunks of 4
#pragma unroll
    for (int s4 = 0; s4 < 4; ++s4) {
      acc0 = wmma_f32(ap[s4], vb0[s4], acc0);
      acc1 = wmma_f32(ap[s4], vb1[s4], acc1);
    }
    __syncthreads();
  }

  float* orow = o + (size_t)(b * SEQ + qs0) * LATD + h * HDIM;
#pragma unroll
  for (int r = 0; r < 8; ++r) {
    const float inv = 1.f / lrow[r];
    float* p = orow + (size_t)(r + 8 * half) * LATD + lm;
    p[0] = acc0[r] * inv;
    p[16] = acc1[r] * inv;
  }
}

// ---------------------------------------------------------------------------

extern "C" void kernel_launch(void* const* d_in, const int* in_sizes, int n_in,
                              void* d_out, int out_size, void* d_ws,
                              size_t ws_size, hipStream_t stream) {
  (void)in_sizes;
  (void)n_in;
  (void)out_size;
  (void)ws_size;
  const float* x = (const float*)d_in[0];
  const float* w_q = (const float*)d_in[1];
  const float* w_k = (const float*)d_in[2];
  const float* w_v = (const float*)d_in[3];
  const float* g_latent = (const float*)d_in[4];
  const float* g_kv = (const float*)d_in[5];
  const float* conv_q_w = (const float*)d_in[6];
  const float* conv_k_w = (const float*)d_in[7];
  const float* g_conv = (const float*)d_in[8];
  const float* g_kconv = (const float*)d_in[9];
  const float* g_postq = (const float*)d_in[10];
  const float* g_postk = (const float*)d_in[11];
  const float* key_temp = (const float*)d_in[12];
  const float* g_preout = (const float*)d_in[13];
  const float* w_o = (const float*)d_in[14];
  float* out = (float*)d_out;

  float* qb = (float*)d_ws;                   // ROWS*LAT : q_pre (in-place rms)
  float* kb = qb + (size_t)ROWS * LATD;       // ROWS*KVD : k_pre
  float* vb = kb + (size_t)ROWS * KVDIM;      // ROWS*KVD : v
  float* qc = vb + (size_t)ROWS * KVDIM;      // ROWS*LAT : q conv -> q final
  float* kc = qc + (size_t)ROWS * LATD;       // ROWS*KVD : k conv -> k final
  float* ao = kc + (size_t)ROWS * KVDIM;      // ROWS*LAT : attn out

  // Projections (fp32 WMMA)
  gemm_nt<<<dim3(LATD / 64, ROWS / 32), 32, 0, stream>>>(x, w_q, qb, LATD, DIMX);
  gemm_nt<<<dim3(KVDIM / 64, ROWS / 32), 32, 0, stream>>>(x, w_k, kb, KVDIM, DIMX);
  gemm_nt<<<dim3(KVDIM / 64, ROWS / 32), 32, 0, stream>>>(x, w_v, vb, KVDIM, DIMX);

  // First RMS norms (in-place) -> q_pre, k_pre, v
  rms_rows<<<ROWS, 32, 0, stream>>>(qb, LATD, g_latent, qb, LATD);
  rms_rows<<<ROWS, 32, 0, stream>>>(kb, KVDIM, g_kv, kb, KVDIM);
  rms_rows<<<ROWS, 32, 0, stream>>>(vb, KVDIM, g_kv, vb, KVDIM);

  // Causal grouped convs + RMS
  conv_causal<<<(ROWS * LATD) / 256, 256, 0, stream>>>(qb, conv_q_w, qc, LATD,
                                                       ROWS * LATD);
  conv_causal<<<(ROWS * KVDIM) / 256, 256, 0, stream>>>(kb, conv_k_w, kc, KVDIM,
                                                        ROWS * KVDIM);
  rms_rows<<<ROWS, 32, 0, stream>>>(qc, LATD, g_conv, qc, LATD);
  rms_rows<<<ROWS, 32, 0, stream>>>(kc, KVDIM, g_kconv, kc, KVDIM);

  // Mixing + post RMS + per-head L2 + RoPE (in-place)
  finalize_q<<<ROWS, 512, 0, stream>>>(qc, kb, g_postq, qc);
  finalize_k<<<ROWS, 128, 0, stream>>>(kc, qb, g_postk, key_temp, kc);

  // Flash attention (fp32 WMMA)
  attn_kernel<<<dim3(SEQ / 16, NHQ, BDIM), 32, 0, stream>>>(qc, kc, vb, ao);

  // Pre-output RMS + output projection
  rms_rows<<<ROWS, 32, 0, stream>>>(ao, LATD, g_preout, ao, LATD);
  gemm_nt<<<dim3(DIMX / 64, ROWS / 32), 32, 0, stream>>>(ao, w_o, out, DIMX,
                                                         LATD);
}